// EnzymeGCN_61804579389955
// MI455X (gfx1250) — compile-verified
//
#include <hip/hip_runtime.h>
#include <hip/hip_bf16.h>
#include <math.h>

typedef __attribute__((ext_vector_type(2))) float v2f;
typedef __attribute__((ext_vector_type(8))) float v8f;

#define HID 64
#define NGRAPH 128
#define NCLS 6

// ---------------------------------------------------------------- utilities
__global__ void zero_kernel(float* __restrict__ p, int n) {
    int i = blockIdx.x * blockDim.x + threadIdx.x;
    if (i < n) p[i] = 0.0f;
}

// deg[dst[e]] += 1 for every edge (self loops folded into dis kernel)
__global__ void deg_kernel(const int* __restrict__ dst, float* __restrict__ deg, int E) {
    int e = blockIdx.x * blockDim.x + threadIdx.x;
    if (e < E) unsafeAtomicAdd(&deg[dst[e]], 1.0f);
}

// dis = rsqrt(deg + 1)   (the +1 is the self loop; deg+1 >= 1 so no max needed)
__global__ void dis_kernel(const float* __restrict__ deg, float* __restrict__ dis, int n) {
    int i = blockIdx.x * blockDim.x + threadIdx.x;
    if (i < n) dis[i] = rsqrtf(deg[i] + 1.0f);
}

// layer-1 linear (K=3) fused with dis pre-scale: hs[i,j] = (x[i,:]@W1[:,j]) * dis[i]
__global__ void gemm1_kernel(const float* __restrict__ x, const float* __restrict__ W1,
                             const float* __restrict__ dis, float* __restrict__ hs, int n) {
    int t = blockIdx.x * blockDim.x + threadIdx.x;
    if (t >= n * HID) return;
    int i = t >> 6;
    int j = t & 63;
    float v = x[i * 3 + 0] * W1[0 * HID + j]
            + x[i * 3 + 1] * W1[1 * HID + j]
            + x[i * 3 + 2] * W1[2 * HID + j];
    hs[t] = v * dis[i];
}

// per-edge scatter: agg[dst, :] += hs[src, :]  (float4 gather, 4 fp32 atomics)
__global__ void scatter_kernel(const int* __restrict__ src, const int* __restrict__ dst,
                               const float* __restrict__ hs, float* __restrict__ agg, int E) {
    int t = blockIdx.x * blockDim.x + threadIdx.x;
    if (t >= E * 16) return;
    int e = t >> 4;          // edge
    int j = (t & 15) << 2;   // feature chunk (4 floats)
    int s = src[e], d = dst[e];
    const float4 m = *reinterpret_cast<const float4*>(&hs[s * HID + j]);
    float* a = &agg[d * HID + j];
    unsafeAtomicAdd(a + 0, m.x);
    unsafeAtomicAdd(a + 1, m.y);
    unsafeAtomicAdd(a + 2, m.z);
    unsafeAtomicAdd(a + 3, m.w);
}

// h[i,j] = relu(dis[i]*(agg[i,j] + hs[i,j]) + b[j])   (self loop = hs[i,j] term)
__global__ void finalize_kernel(const float* __restrict__ agg, const float* __restrict__ hs,
                                const float* __restrict__ dis, const float* __restrict__ b,
                                float* __restrict__ h, int n) {
    int t = blockIdx.x * blockDim.x + threadIdx.x;
    if (t >= n * HID) return;
    int i = t >> 6;
    int j = t & 63;
    float v = dis[i] * (agg[t] + hs[t]) + b[j];
    h[t] = v > 0.0f ? v : 0.0f;
}

// ------------------------------------------------- layer-2 GEMM via fp32 WMMA
// One wave (32 lanes) computes a 16-row x 64-col stripe of h1 @ W2, K=64 as
// 16 chained V_WMMA_F32_16X16X4_F32 per 16-wide N tile. Epilogue fuses *dis.
__global__ __launch_bounds__(32)
void gemm2_wmma_kernel(const float* __restrict__ h1, const float* __restrict__ W2,
                       const float* __restrict__ dis, float* __restrict__ hs2, int nrows) {
    const int lane = threadIdx.x;      // 0..31
    const int m    = lane & 15;        // row-in-tile (A) / col-in-tile (B,C)
    const int hi   = lane >> 4;        // lane-half selector
    const int row0 = blockIdx.x * 16;
    int ar = row0 + m;                 // A row for this lane (clamp keeps EXEC full)
    if (ar >= nrows) ar = nrows - 1;

    for (int nt = 0; nt < 4; ++nt) {
        v8f c = {};
        const int ncol = nt * 16 + m;
        #pragma unroll
        for (int k0 = 0; k0 < HID; k0 += 4) {
            // A 16x4 layout: lanes 0-15 hold K=k0,k0+1; lanes 16-31 hold K=k0+2,k0+3
            const int ka = k0 + hi * 2;
            v2f a, b;
            a.x = h1[ar * HID + ka];
            a.y = h1[ar * HID + ka + 1];
            // B 4x16 layout: VGPR0 = rows K=k0 (lo lanes) / k0+2 (hi lanes); VGPR1 = k0+1 / k0+3
            b.x = W2[(k0 + hi * 2) * HID + ncol];
            b.y = W2[(k0 + hi * 2 + 1) * HID + ncol];
            c = __builtin_amdgcn_wmma_f32_16x16x4_f32(
                    false, a, false, b, (short)0, c, false, false);
        }
        // C/D layout: VGPR v -> row (v + hi*8), col = m (within tile)
        #pragma unroll
        for (int v = 0; v < 8; ++v) {
            const int mr = row0 + v + hi * 8;
            if (mr < nrows) hs2[mr * HID + nt * 16 + m] = c[v] * dis[mr];
        }
    }
}

// finalize layer 2 + global mean-pool accumulation
__global__ void finalize_pool_kernel(const float* __restrict__ agg, const float* __restrict__ hs,
                                     const float* __restrict__ dis, const float* __restrict__ b,
                                     const int* __restrict__ batch, float* __restrict__ pooled,
                                     float* __restrict__ cnt, int n) {
    int t = blockIdx.x * blockDim.x + threadIdx.x;
    if (t >= n * HID) return;
    int i = t >> 6;
    int j = t & 63;
    float v = dis[i] * (agg[t] + hs[t]) + b[j];
    v = v > 0.0f ? v : 0.0f;
    int g = batch[i];
    unsafeAtomicAdd(&pooled[g * HID + j], v);
    if (j == 0) unsafeAtomicAdd(&cnt[g], 1.0f);
}

// head: mean, linear (64x6), log_softmax; one thread per graph
__global__ void head_kernel(const float* __restrict__ pooled, const float* __restrict__ cnt,
                            const float* __restrict__ Wl, const float* __restrict__ bl,
                            float* __restrict__ out) {
    int g = blockIdx.x * blockDim.x + threadIdx.x;
    if (g >= NGRAPH) return;
    float inv = 1.0f / fmaxf(cnt[g], 1.0f);
    float s[NCLS];
    #pragma unroll
    for (int c = 0; c < NCLS; ++c) s[c] = bl[c];
    for (int j = 0; j < HID; ++j) {
        float p = pooled[g * HID + j] * inv;
        #pragma unroll
        for (int c = 0; c < NCLS; ++c) s[c] += p * Wl[j * NCLS + c];
    }
    float mx = s[0];
    #pragma unroll
    for (int c = 1; c < NCLS; ++c) mx = fmaxf(mx, s[c]);
    float lse = 0.0f;
    #pragma unroll
    for (int c = 0; c < NCLS; ++c) lse += expf(s[c] - mx);
    lse = logf(lse);
    #pragma unroll
    for (int c = 0; c < NCLS; ++c) out[g * NCLS + c] = s[c] - mx - lse;
}

// ---------------------------------------------------------------- launcher
extern "C" void kernel_launch(void* const* d_in, const int* in_sizes, int n_in,
                              void* d_out, int out_size, void* d_ws, size_t ws_size,
                              hipStream_t stream) {
    const float* x     = (const float*)d_in[0];
    const int*   eidx  = (const int*)  d_in[1];
    const int*   batch = (const int*)  d_in[2];
    const float* W1    = (const float*)d_in[3];
    const float* b1    = (const float*)d_in[4];
    const float* W2    = (const float*)d_in[5];
    const float* b2    = (const float*)d_in[6];
    const float* Wl    = (const float*)d_in[7];
    const float* bl    = (const float*)d_in[8];
    float* out = (float*)d_out;

    const int NN = in_sizes[0] / 3;       // 100000 nodes
    const int E  = in_sizes[1] / 2;       // 3200000 edges
    const int* src = eidx;                // edge_index[0]
    const int* dst = eidx + E;            // edge_index[1]

    // workspace layout (floats)
    float* ws     = (float*)d_ws;
    float* deg    = ws;                       // NN
    float* dis    = deg + NN;                 // NN
    float* hs     = dis + NN;                 // NN*HID  (messages, reused per layer)
    float* agg    = hs + (size_t)NN * HID;    // NN*HID  (scatter target, reused)
    float* h1     = agg + (size_t)NN * HID;   // NN*HID
    float* pooled = h1 + (size_t)NN * HID;    // NGRAPH*HID
    float* cnt    = pooled + NGRAPH * HID;    // NGRAPH

    const int B = 256;
    const int nNF = NN * HID;
    #define GRID(n) (((n) + B - 1) / B)

    // zero accumulators
    zero_kernel<<<GRID(NN), B, 0, stream>>>(deg, NN);
    zero_kernel<<<GRID(nNF), B, 0, stream>>>(agg, nNF);
    zero_kernel<<<GRID(NGRAPH * HID + NGRAPH), B, 0, stream>>>(pooled, NGRAPH * HID + NGRAPH);

    // degree + normalization
    deg_kernel<<<GRID(E), B, 0, stream>>>(dst, deg, E);
    dis_kernel<<<GRID(NN), B, 0, stream>>>(deg, dis, NN);

    // layer 1
    gemm1_kernel<<<GRID(nNF), B, 0, stream>>>(x, W1, dis, hs, NN);
    scatter_kernel<<<GRID(E * 16), B, 0, stream>>>(src, dst, hs, agg, E);
    finalize_kernel<<<GRID(nNF), B, 0, stream>>>(agg, hs, dis, b1, h1, NN);

    // layer 2
    zero_kernel<<<GRID(nNF), B, 0, stream>>>(agg, nNF);
    gemm2_wmma_kernel<<<(NN + 15) / 16, 32, 0, stream>>>(h1, W2, dis, hs, NN);
    scatter_kernel<<<GRID(E * 16), B, 0, stream>>>(src, dst, hs, agg, E);
    finalize_pool_kernel<<<GRID(nNF), B, 0, stream>>>(agg, hs, dis, b2, batch, pooled, cnt, NN);

    // head
    head_kernel<<<1, NGRAPH, 0, stream>>>(pooled, cnt, Wl, bl, out);
    #undef GRID
}